// SectorGNN_25821343383879
// MI455X (gfx1250) — compile-verified
//
#include <hip/hip_runtime.h>
#include <hip/hip_bf16.h>

typedef __attribute__((ext_vector_type(2))) float v2f;
typedef __attribute__((ext_vector_type(8))) float v8f;

#define F_IN 256
#define HID  32

// ---------------- degree / normalization ----------------

__global__ void k_deg_init(float* __restrict__ deg, int n) {
    int i = blockIdx.x * blockDim.x + threadIdx.x;
    if (i < n) deg[i] = 1.0f;  // self-loop contributes 1
}

__global__ void k_deg_edges(const int* __restrict__ ei, int E, float* __restrict__ deg) {
    int e = blockIdx.x * blockDim.x + threadIdx.x;
    if (e < E) atomicAdd(&deg[ei[E + e]], 1.0f);  // dst = ei[1][e]
}

__global__ void k_dinv(const float* __restrict__ deg, float* __restrict__ dinv, int n) {
    int i = blockIdx.x * blockDim.x + threadIdx.x;
    if (i < n) {
        float d = deg[i];
        dinv[i] = (d > 0.0f) ? rsqrtf(d) : 0.0f;
    }
}

// ---------------- GEMM1: h = x @ W1  via V_WMMA_F32_16X16X4_F32 ----------------
// Block = 64 threads = 2 waves. Block computes a 16-row tile x full 32 cols.
// wave 0 -> cols 0..15, wave 1 -> cols 16..31. N is a multiple of 16 (100000),
// so no tail guards are needed and EXEC stays all-1s around the WMMAs.

__global__ __launch_bounds__(64) void k_gemm1_wmma(const float* __restrict__ x,
                                                   const float* __restrict__ W1,
                                                   float* __restrict__ h, int n) {
    (void)n;
    __shared__ __align__(16) float As[16 * 260];   // 16 x 256, stride 260 (260%64==4 -> conflict-free)
    __shared__ __align__(16) float Bs[F_IN * HID]; // 256 x 32 row-major (k, n)

    const int t       = threadIdx.x;
    const int rowBase = blockIdx.x * 16;

    // Stage A tile (16x256 f32) via CDNA5 async DMA to LDS
    // (GLOBAL_LOAD_ASYNC_TO_LDS_B128, tracked by ASYNCcnt; no VGPR round-trip).
    // unroll 1: keep address math incremental so we don't pin ~150 VGPRs of
    // hoisted addresses — occupancy is what hides the DMA latency.
    // With 64 threads: slot (i, t) -> row i, float4-col t.
#pragma unroll 1
    for (int i = 0; i < 16; ++i) {
        const float* gp = x + (size_t)(rowBase + i) * F_IN + t * 4;
        unsigned lds_off = (unsigned)(unsigned long long)(const void*)&As[i * 260 + t * 4];
        unsigned long long ga = (unsigned long long)gp;
        asm volatile("global_load_async_to_lds_b128 %0, %1, off"
                     :: "v"(lds_off), "v"(ga) : "memory");
    }
    // Stage W1 (256x32 f32 = 2048 float4) the same way.
#pragma unroll 1
    for (int i = 0; i < 32; ++i) {
        int idx = t + i * 64;              // 0..2047
        const float* gp = W1 + (size_t)idx * 4;
        unsigned lds_off = (unsigned)(unsigned long long)(const void*)&Bs[idx * 4];
        unsigned long long ga = (unsigned long long)gp;
        asm volatile("global_load_async_to_lds_b128 %0, %1, off"
                     :: "v"(lds_off), "v"(ga) : "memory");
    }
    asm volatile("s_wait_asynccnt 0x0" ::: "memory");
    __syncthreads();

    const int lane = t & 31;
    const int wn   = t >> 5;        // which 16-col tile
    const int half = lane >> 4;     // 0: K pair {0,1}/M 0..7 of C ; 1: K pair {2,3}/M 8..15
    const int mn   = lane & 15;     // M index for A, N index for B/C/D

    v8f c = {};
#pragma unroll 4
    for (int kb = 0; kb < F_IN; kb += 4) {
        const int ka = kb + 2 * half;
        v2f a, b;
        a.x = As[mn * 260 + ka];
        a.y = As[mn * 260 + ka + 1];
        b.x = Bs[ka * HID + wn * 16 + mn];
        b.y = Bs[(ka + 1) * HID + wn * 16 + mn];
        c = __builtin_amdgcn_wmma_f32_16x16x4_f32(
                /*neg_a=*/false, a, /*neg_b=*/false, b,
                /*c_mod=*/(short)0, c, /*reuse_a=*/false, /*reuse_b=*/false);
    }

    // C/D layout: VGPR r -> M = r + 8*half, N = mn. No guards (N % 16 == 0).
#pragma unroll
    for (int r = 0; r < 8; ++r) {
        int row = rowBase + r + 8 * half;
        h[(size_t)row * HID + wn * 16 + mn] = c[r];
    }
}

// ---------------- aggregation layer 1 ----------------

__global__ void k_agg1_init(const float* __restrict__ h, const float* __restrict__ dinv,
                            float* __restrict__ agg, int n) {
    int tid = blockIdx.x * blockDim.x + threadIdx.x;
    int i = tid >> 5, f = tid & 31;
    if (i < n) {
        float di = dinv[i];
        agg[(size_t)i * HID + f] = h[(size_t)i * HID + f] * di * di;  // self-loop msg
    }
}

__global__ void k_scatter1(const int* __restrict__ ei, int E,
                           const float* __restrict__ h, const float* __restrict__ dinv,
                           float* __restrict__ agg) {
    int tid = blockIdx.x * blockDim.x + threadIdx.x;
    int e = tid >> 5, f = tid & 31;
    if (e < E) {
        int s = ei[e], d = ei[E + e];
        float norm = dinv[s] * dinv[d];
        atomicAdd(&agg[(size_t)d * HID + f], h[(size_t)s * HID + f] * norm);
    }
}

// ---------------- layer 2 transform: h2 = relu(agg1 + b1) @ W2 ----------------

__global__ void k_layer2(const float* __restrict__ agg, const float* __restrict__ b1,
                         const float* __restrict__ W2, float* __restrict__ h2, int n) {
    int i = blockIdx.x * blockDim.x + threadIdx.x;
    if (i >= n) return;
    const float4* a4 = (const float4*)(agg + (size_t)i * HID);
    const float4* b4 = (const float4*)b1;
    const float4* w4 = (const float4*)W2;
    float acc = 0.0f;
#pragma unroll
    for (int q = 0; q < HID / 4; ++q) {
        float4 a = a4[q], bb = b4[q], w = w4[q];
        acc += fmaxf(a.x + bb.x, 0.0f) * w.x;
        acc += fmaxf(a.y + bb.y, 0.0f) * w.y;
        acc += fmaxf(a.z + bb.z, 0.0f) * w.z;
        acc += fmaxf(a.w + bb.w, 0.0f) * w.w;
    }
    h2[i] = acc;
}

// ---------------- aggregation layer 2 (scalar features) ----------------

__global__ void k_out_init(const float* __restrict__ h2, const float* __restrict__ dinv,
                           const float* __restrict__ b2, float* __restrict__ out, int n) {
    int i = blockIdx.x * blockDim.x + threadIdx.x;
    if (i < n) {
        float di = dinv[i];
        out[i] = h2[i] * di * di + b2[0];  // self-loop msg + output bias
    }
}

__global__ void k_scatter2(const int* __restrict__ ei, int E,
                           const float* __restrict__ h2, const float* __restrict__ dinv,
                           float* __restrict__ out) {
    int e = blockIdx.x * blockDim.x + threadIdx.x;
    if (e < E) {
        int s = ei[e], d = ei[E + e];
        atomicAdd(&out[d], h2[s] * dinv[s] * dinv[d]);
    }
}

// ---------------- launcher ----------------

extern "C" void kernel_launch(void* const* d_in, const int* in_sizes, int n_in,
                              void* d_out, int out_size, void* d_ws, size_t ws_size,
                              hipStream_t stream) {
    const float* x  = (const float*)d_in[0];
    const int*   ei = (const int*)d_in[1];
    const float* W1 = (const float*)d_in[2];
    const float* b1 = (const float*)d_in[3];
    const float* W2 = (const float*)d_in[4];
    const float* b2 = (const float*)d_in[5];
    float* out = (float*)d_out;

    const int N = in_sizes[0] / F_IN;   // 100000
    const int E = in_sizes[1] / 2;      // 1600000

    // Workspace carve-out: deg, dinv, h [N,32], agg1 [N,32], h2 [N]
    char* p = (char*)d_ws;
    float* deg  = (float*)p; p += (size_t)N * sizeof(float);
    float* dinv = (float*)p; p += (size_t)N * sizeof(float);
    float* h    = (float*)p; p += (size_t)N * HID * sizeof(float);
    float* agg1 = (float*)p; p += (size_t)N * HID * sizeof(float);
    float* h2   = (float*)p;

    const int TB = 256;
    k_deg_init <<<(N + TB - 1) / TB, TB, 0, stream>>>(deg, N);
    k_deg_edges<<<(E + TB - 1) / TB, TB, 0, stream>>>(ei, E, deg);
    k_dinv     <<<(N + TB - 1) / TB, TB, 0, stream>>>(deg, dinv, N);

    k_gemm1_wmma<<<(N + 15) / 16, 64, 0, stream>>>(x, W1, h, N);

    {
        long long tot = (long long)N * HID;
        k_agg1_init<<<(int)((tot + TB - 1) / TB), TB, 0, stream>>>(h, dinv, agg1, N);
    }
    {
        long long tot = (long long)E * HID;
        k_scatter1<<<(int)((tot + TB - 1) / TB), TB, 0, stream>>>(ei, E, h, dinv, agg1);
    }

    k_layer2  <<<(N + TB - 1) / TB, TB, 0, stream>>>(agg1, b1, W2, h2, N);
    k_out_init<<<(N + TB - 1) / TB, TB, 0, stream>>>(h2, dinv, b2, out, N);
    k_scatter2<<<(E + TB - 1) / TB, TB, 0, stream>>>(ei, E, h2, dinv, out);
}